// STLinearAttention_73349451481745
// MI455X (gfx1250) — compile-verified
//
#include <hip/hip_runtime.h>
#include <math.h>
#include <stdint.h>

typedef __attribute__((ext_vector_type(2))) float v2f;
typedef __attribute__((ext_vector_type(8))) float v8f;

#define EPS   1e-6f
#define LL    256     // sequence length (l and s)
#define DD    32      // head dim (d and m)
#define HH    8       // heads
#define KSTR  36      // padded LDS row stride for K/V/Q tiles (16B aligned, bank-spread)
#define KVSTR 33      // padded LDS row stride for KVT

__device__ __forceinline__ float fmap(float x) {
    // elu(x) + 1  ==  x+1 for x>=0, exp(x) for x<0
    return x >= 0.0f ? x + 1.0f : __expf(x);
}

// Input layout [b,n,l,h,d]: elem(bn, s, h, d) = bn*65536 + s*256 + h*32 + d

extern "C" __global__ __launch_bounds__(128)
void linattn_wmma_kernel(const float* __restrict__ qp,
                         const float* __restrict__ kp,
                         const float* __restrict__ vp,
                         float* __restrict__ outp)
{
    extern __shared__ float smem[];
    float* sK    = smem;                   // 256*36
    float* sV    = sK  + LL * KSTR;        // 256*36
    float* sKVT  = sV  + LL * KSTR;        // 32*33   (KVT[d][m])
    float* sKsum = sKVT + 32 * KVSTR;      // 32
    float* sQ    = sKsum + 32;             // 4 waves * 16*36
    float* sZ    = sQ + 4 * 16 * KSTR;     // 4 waves * 16

    const int tid  = threadIdx.x;
    const int lane = tid & 31;
    const int wave = tid >> 5;

    const int g  = blockIdx.x;             // 2048 groups = bn*8 + h
    const int h  = g & (HH - 1);
    const int bn = g >> 3;
    const size_t base = (size_t)bn * 65536 + (size_t)h * 32;

    // ---------- Phase A0: V -> LDS via CDNA5 async global->LDS DMA (no transform needed) ----------
    // Each lane issues b128 async copies; tracked by ASYNCcnt, drained before the barrier.
    for (int idx = tid; idx < LL * 8; idx += 128) {
        const int s  = idx >> 3;
        const int qd = (idx & 7) * 4;
        const float* gsrc = vp + base + (size_t)s * 256 + qd;
        const uint32_t ldst = (uint32_t)(uintptr_t)(sV + s * KSTR + qd); // LDS byte addr = flat[31:0]
        asm volatile("global_load_async_to_lds_b128 %0, %1, off"
                     :: "v"(ldst), "v"(gsrc) : "memory");
    }

    // ---------- Phase A1: stage fmap(K) into LDS (regular path, transform in VALU) ----------
    for (int idx = tid; idx < LL * 8; idx += 128) {
        const int s  = idx >> 3;
        const int qd = (idx & 7) * 4;
        float4 kk = *(const float4*)(kp + base + (size_t)s * 256 + qd);
        kk.x = fmap(kk.x); kk.y = fmap(kk.y); kk.z = fmap(kk.z); kk.w = fmap(kk.w);
        *(float4*)(sK + s * KSTR + qd) = kk;
    }

    asm volatile("s_wait_asynccnt 0x0" ::: "memory");  // V DMA complete before barrier
    __syncthreads();

    // ---------- Phase B: Ksum (wave 0 lanes, overlapped with WMMA below) ----------
    if (tid < 32) {
        float acc = 0.0f;
        for (int s = 0; s < LL; ++s) acc += sK[s * KSTR + tid];
        sKsum[tid] = acc;
    }

    // ---------- Phase C: KVT = K^T @ V  (one 16x16 tile per wave, K-loop over s) ----------
    {
        const int dt   = wave >> 1;        // d-tile
        const int mt   = wave & 1;         // m-tile
        const int half = lane >> 4;
        const int lid  = lane & 15;
        const int dd   = dt * 16 + lid;    // A row (d)
        const int mm   = mt * 16 + lid;    // B col (m)
        v8f c = {};
        for (int s0 = 0; s0 < LL; s0 += 4) {
            const int s = s0 + 2 * half;
            v2f a, b;
            a.x = sK[(s    ) * KSTR + dd];
            a.y = sK[(s + 1) * KSTR + dd];
            b.x = sV[(s    ) * KSTR + mm];
            b.y = sV[(s + 1) * KSTR + mm];
            c = __builtin_amdgcn_wmma_f32_16x16x4_f32(false, a, false, b,
                                                      (short)0, c, false, false);
        }
#pragma unroll
        for (int r = 0; r < 8; ++r)
            sKVT[(dt * 16 + half * 8 + r) * KVSTR + mm] = c[r];
    }
    __syncthreads();

    // ---------- Phase D: Out = Z * (fmap(Q) @ KVT) ----------
    const int half = lane >> 4;
    const int lid  = lane & 15;

    // Preload all B fragments (KVT) once: 8 k-steps x 2 m-tiles (32 VGPRs)
    v2f bf[8][2];
#pragma unroll
    for (int ks = 0; ks < 8; ++ks) {
        const int d0 = ks * 4 + 2 * half;
#pragma unroll
        for (int mt = 0; mt < 2; ++mt) {
            bf[ks][mt].x = sKVT[(d0    ) * KVSTR + mt * 16 + lid];
            bf[ks][mt].y = sKVT[(d0 + 1) * KVSTR + mt * 16 + lid];
        }
    }

    float* sQw = sQ + wave * 16 * KSTR;
    float* sZw = sZ + wave * 16;

    for (int i = 0; i < 4; ++i) {
        const int lb = (wave * 4 + i) * 16;      // l-tile base row
        // stage this wave's fmap(Q) tile (coalesced float4)
#pragma unroll
        for (int t = 0; t < 4; ++t) {
            const int idx = t * 32 + lane;
            const int row = idx >> 3;
            const int qd  = (idx & 7) * 4;
            float4 qq = *(const float4*)(qp + base + (size_t)(lb + row) * 256 + qd);
            qq.x = fmap(qq.x); qq.y = fmap(qq.y); qq.z = fmap(qq.z); qq.w = fmap(qq.w);
            *(float4*)(sQw + row * KSTR + qd) = qq;
        }
        __syncthreads();
        // Z for the 16 rows of this tile
        if (lane < 16) {
            float acc = 0.0f;
#pragma unroll
            for (int d = 0; d < DD; ++d) acc += sQw[lane * KSTR + d] * sKsum[d];
            sZw[lane] = 1.0f / (acc + EPS);
        }
        __syncthreads();

        v8f c0 = {}, c1 = {};
#pragma unroll
        for (int ks = 0; ks < 8; ++ks) {
            const int d0 = ks * 4 + 2 * half;
            v2f a = *(const v2f*)(sQw + lid * KSTR + d0);
            c0 = __builtin_amdgcn_wmma_f32_16x16x4_f32(false, a, false, bf[ks][0],
                                                       (short)0, c0, false, false);
            c1 = __builtin_amdgcn_wmma_f32_16x16x4_f32(false, a, false, bf[ks][1],
                                                       (short)0, c1, false, false);
        }
        // scale by Z and store
#pragma unroll
        for (int r = 0; r < 8; ++r) {
            const int row = half * 8 + r;
            const float z = sZw[row];
            const size_t o = base + (size_t)(lb + row) * 256;
            outp[o + lid]      = c0[r] * z;
            outp[o + 16 + lid] = c1[r] * z;
        }
    }
}

extern "C" void kernel_launch(void* const* d_in, const int* in_sizes, int n_in,
                              void* d_out, int out_size, void* d_ws, size_t ws_size,
                              hipStream_t stream) {
    const float* q = (const float*)d_in[0];
    const float* k = (const float*)d_in[1];
    const float* v = (const float*)d_in[2];
    float* out = (float*)d_out;
    const size_t shmem =
        (size_t)(2 * LL * KSTR + 32 * KVSTR + 32 + 4 * 16 * KSTR + 4 * 16) * sizeof(float);
    linattn_wmma_kernel<<<2048, 128, shmem, stream>>>(q, k, v, out);
}